// OperationScorer_35536559407343
// MI455X (gfx1250) — compile-verified
//
#include <hip/hip_runtime.h>
#include <hip/hip_bf16.h>

typedef __attribute__((ext_vector_type(2))) float v2f;
typedef __attribute__((ext_vector_type(8))) float v8f;

#define RDIM 2048   // B*T = B*S rows
#define KDIM 512    // M = N (reduction dim)
#define HDIM 128
#define INV_SCALE 0.04419417382415922f  // 1/sqrt(512)

// One wave (32 threads) per (16-row tile, side). Each wave computes the full
// H=128 projection for its 16 rows using 8 f32-WMMA accumulators (one per
// 16-wide h tile), then folds tanh()*v over all 128 h in-register + 4 shuffle
// steps, writing the final per-row scalar.
//   side 0: a[row] = sum_h v[h]     * tanh( (query . W2^T)[row,h] )
//   side 1: c[row] = sum_h v[128+h] * tanh( (keys  . W1^T)[row,h] )
__global__ __launch_bounds__(32) void proj_tanh_dot_kernel(
    const float* __restrict__ query,
    const float* __restrict__ keys,
    const float* __restrict__ W1,
    const float* __restrict__ W2,
    const float* __restrict__ v,
    float* __restrict__ avec,    // [2048]
    float* __restrict__ cvec)    // [2048]
{
    const int sel  = blockIdx.z;
    const float* __restrict__ A = sel ? keys : query;
    const float* __restrict__ W = sel ? W1 : W2;          // W2 pairs with query
    const float* __restrict__ vseg = v + (sel ? HDIM : 0);
    float* __restrict__ outP = sel ? cvec : avec;

    const int lane = threadIdx.x;
    const int m    = lane & 15;   // A: row-in-tile / B: h-in-tile
    const int half = lane >> 4;   // selects K pair within each K=4 step

    const int rowBase = blockIdx.x * 16;

    // A fragment (16x4 f32): lane holds A[rowBase+m][k + half*2 + {0,1}]
    const float* aPtr = A + (size_t)(rowBase + m) * KDIM + half * 2;
    // B fragment (4x16 f32) = W^T tile ht: lane holds W[ht*16+m][k + half*2 + {0,1}]
    const float* wPtr = W + (size_t)m * KDIM + half * 2;

    v8f acc[8];
#pragma unroll
    for (int ht = 0; ht < 8; ++ht)
        acc[ht] = (v8f){0.f, 0.f, 0.f, 0.f, 0.f, 0.f, 0.f, 0.f};

#pragma unroll 2
    for (int k = 0; k < KDIM; k += 4) {
        v2f av = *(const v2f*)(aPtr + k);               // 1 x global_load_b64
        v2f wv[8];
#pragma unroll
        for (int ht = 0; ht < 8; ++ht)                   // 8 x global_load_b64
            wv[ht] = *(const v2f*)(wPtr + k + ht * (16 * KDIM));
#pragma unroll
        for (int ht = 0; ht < 8; ++ht)
            acc[ht] = __builtin_amdgcn_wmma_f32_16x16x4_f32(
                /*neg_a=*/false, av, /*neg_b=*/false, wv[ht],
                /*c_mod=*/(short)0, acc[ht],
                /*reuse_a=*/false, /*reuse_b=*/false);
    }

    // acc[ht][r] = x[row = rowBase + r + half*8][h = ht*16 + m]
#pragma unroll
    for (int r = 0; r < 8; ++r) {
        float t = 0.f;
#pragma unroll
        for (int ht = 0; ht < 8; ++ht)
            t += tanhf(acc[ht][r]) * vseg[ht * 16 + m];
        // reduce across the 16 lanes of this half-wave (all m within h tile)
        t += __shfl_xor(t, 1, 32);
        t += __shfl_xor(t, 2, 32);
        t += __shfl_xor(t, 4, 32);
        t += __shfl_xor(t, 8, 32);
        if (m == 0) {
            outP[rowBase + r + half * 8] = t;
        }
    }
}

// out[b,t,s] = (a[b*512+t] + c[b*512+s]) * INV_SCALE ; float4 stores along s.
__global__ __launch_bounds__(256) void broadcast_sum_kernel(
    const float* __restrict__ a, const float* __restrict__ c,
    float* __restrict__ out)
{
    int idx = blockIdx.x * blockDim.x + threadIdx.x;   // 0 .. 262143
    int s4  = (idx & 127) * 4;                          // 512/4 groups per row
    int bt  = idx >> 7;                                 // b*512 + t
    int b   = bt >> 9;

    float av = a[bt];
    const float* crow = c + b * 512;

    float4 o;
    o.x = (av + crow[s4 + 0]) * INV_SCALE;
    o.y = (av + crow[s4 + 1]) * INV_SCALE;
    o.z = (av + crow[s4 + 2]) * INV_SCALE;
    o.w = (av + crow[s4 + 3]) * INV_SCALE;
    *(float4*)(out + (size_t)bt * 512 + s4) = o;
}

extern "C" void kernel_launch(void* const* d_in, const int* in_sizes, int n_in,
                              void* d_out, int out_size, void* d_ws, size_t ws_size,
                              hipStream_t stream) {
    const float* query = (const float*)d_in[0];   // (4,512,512)
    const float* keys  = (const float*)d_in[1];   // (4,512,512)
    const float* W1    = (const float*)d_in[2];   // (128,512)
    const float* W2    = (const float*)d_in[3];   // (128,512)
    const float* v     = (const float*)d_in[4];   // (1,256)
    float* out = (float*)d_out;                   // (4,512,512)

    float* wsf  = (float*)d_ws;
    float* avec = wsf;              // 2048 floats
    float* cvec = wsf + RDIM;       // 2048 floats  (total ws use: 16 KB)

    dim3 g1(RDIM / 16, 1, 2);       // 128 row-tiles x 2 sides = 256 waves
    proj_tanh_dot_kernel<<<g1, dim3(32), 0, stream>>>(query, keys, W1, W2, v,
                                                      avec, cvec);

    // 4*512*512 outputs / 4 per thread = 262144 threads
    broadcast_sum_kernel<<<262144 / 256, 256, 0, stream>>>(avec, cvec, out);
}